// SentimentAnalysisModel_87771951661695
// MI455X (gfx1250) — compile-verified
//
#include <hip/hip_runtime.h>
#include <hip/hip_bf16.h>
#include <math.h>

// ---------------- problem constants ----------------
#define N_NODES  131072
#define N_EDGES  1048576
#define N_GRAPHS 1024
#define EMB      96
#define HID      64
#define OUTC     3
#define ET       (N_EDGES + N_NODES)   // edges incl. self loops
#define NEG_SLOPE 0.2f

// ---------------- workspace layout (float offsets) ----------------
#define OFF_H   0ull                                   // [N, HID]  h = x@W
#define OFF_AS  (OFF_H  + (size_t)N_NODES * HID)       // [N] a_src
#define OFF_AD  (OFF_AS + (size_t)N_NODES)             // [N] a_dst
#define OFF_M   (OFF_AD + (size_t)N_NODES)             // [N] ordered-uint max
#define OFF_Z   (OFF_M  + (size_t)N_NODES)             // [N] softmax denom
#define OFF_E   (OFF_Z  + (size_t)N_NODES)             // [ET] logit -> exp
#define OFF_O   (OFF_E  + (size_t)ET)                  // [N, HID] aggregated out
#define OFF_P   (OFF_O  + (size_t)N_NODES * HID)       // [G, HID] pooled sums
#define OFF_C   (OFF_P  + (size_t)N_GRAPHS * HID)      // [G] counts
#define WS_FLOATS (OFF_C + (size_t)N_GRAPHS)

typedef float v2f __attribute__((ext_vector_type(2)));
typedef float v8f __attribute__((ext_vector_type(8)));

// monotonic float <-> ordered uint mapping (for atomic max on floats)
__device__ __forceinline__ unsigned int fl_flip(float x) {
    unsigned int f = __float_as_uint(x);
    unsigned int mask = (unsigned int)(-(int)(f >> 31)) | 0x80000000u;
    return f ^ mask;
}
__device__ __forceinline__ float fl_unflip(unsigned int f) {
    unsigned int mask = ((f >> 31) - 1u) | 0x80000000u;
    return __uint_as_float(f ^ mask);
}

// ================= K1: h = x @ W via V_WMMA_F32_16X16X4_F32 =================
// Grid: 1024 blocks x 256 threads (8 waves). Wave w computes rows
// [(bid*8+w)*16, +16) x all 64 cols as four 16x16 f32 accumulators.
#define LDS_STRIDE 98   // even (b64 alignment), odd/2 mod 64 -> conflict-light
__global__ __launch_bounds__(256)
void gat_gemm_wmma(const float* __restrict__ x, const float* __restrict__ W,
                   float* __restrict__ h) {
    __shared__ float Wt[HID * LDS_STRIDE];            // W transposed: Wt[c][k]
    // cooperative load of W (EMB*HID = 6144 floats)
    for (int idx = threadIdx.x; idx < EMB * HID; idx += 256) {
        int k = idx / HID, c = idx % HID;
        Wt[c * LDS_STRIDE + k] = W[idx];
    }
    __syncthreads();

    const int wave = threadIdx.x >> 5;
    const int lane = threadIdx.x & 31;
    const int rowBase = (blockIdx.x * 8 + wave) * 16;
    const int l15 = lane & 15;
    const int kHalf = (lane >> 4) << 1;               // 0 or 2

    v8f acc[4] = {};                                  // 4 col-tiles of 16x16

    const float* xrow = x + (size_t)(rowBase + l15) * EMB;
    for (int k = 0; k < EMB; k += 4) {
        // A fragment: 16x4, lanes 0-15 hold K=k,k+1 ; lanes 16-31 K=k+2,k+3
        v2f a = *(const v2f*)(xrow + k + kHalf);
#pragma unroll
        for (int c = 0; c < 4; ++c) {
            int col = c * 16 + l15;
            // B fragment: 4x16 mirrored layout, contiguous pair in LDS
            v2f b = *(const v2f*)(&Wt[col * LDS_STRIDE + k + kHalf]);
            acc[c] = __builtin_amdgcn_wmma_f32_16x16x4_f32(
                false, a, false, b, (short)0, acc[c], false, false);
        }
    }

    // C/D layout: VGPR g, lanes 0-15 -> (M=g, N=lane); lanes 16-31 -> (M=g+8)
    const int rOff = rowBase + ((lane >> 4) << 3);
#pragma unroll
    for (int c = 0; c < 4; ++c) {
        int col = c * 16 + l15;
#pragma unroll
        for (int g = 0; g < 8; ++g)
            h[(size_t)(rOff + g) * HID + col] = acc[c][g];
    }
}

// ================= K2: a_s = h.att_src, a_d = h.att_dst (wave/node) =========
__global__ __launch_bounds__(256)
void gat_attn_dots(const float* __restrict__ h, const float* __restrict__ att_src,
                   const float* __restrict__ att_dst,
                   float* __restrict__ a_s, float* __restrict__ a_d) {
    int n = (blockIdx.x * blockDim.x + threadIdx.x) >> 5;
    int lane = threadIdx.x & 31;
    if (n >= N_NODES) return;
    float2 hv = *(const float2*)(h + (size_t)n * HID + lane * 2);
    float2 s2 = *(const float2*)(att_src + lane * 2);
    float2 d2 = *(const float2*)(att_dst + lane * 2);
    float ps = hv.x * s2.x + hv.y * s2.y;
    float pd = hv.x * d2.x + hv.y * d2.y;
#pragma unroll
    for (int off = 16; off >= 1; off >>= 1) {
        ps += __shfl_xor(ps, off, 32);
        pd += __shfl_xor(pd, off, 32);
    }
    if (lane == 0) { a_s[n] = ps; a_d[n] = pd; }
}

// ================= K3: per-edge logit + segment max =========================
__global__ __launch_bounds__(256)
void gat_edge_logits(const int* __restrict__ esrc, const int* __restrict__ edst,
                     const float* __restrict__ a_s, const float* __restrict__ a_d,
                     float* __restrict__ ebuf, unsigned int* __restrict__ mmax) {
    int i = blockIdx.x * blockDim.x + threadIdx.x;
    if (i >= ET) return;
    int s, d;
    if (i < N_EDGES) { s = esrc[i]; d = edst[i]; } else { s = d = i - N_EDGES; }
    float l = a_s[s] + a_d[d];
    l = l > 0.0f ? l : NEG_SLOPE * l;
    ebuf[i] = l;
    atomicMax(&mmax[d], fl_flip(l));
}

// ================= K4: exp(logit - max) + segment sum =======================
__global__ __launch_bounds__(256)
void gat_edge_exp(const int* __restrict__ edst, const unsigned int* __restrict__ mmax,
                  float* __restrict__ ebuf, float* __restrict__ z) {
    int i = blockIdx.x * blockDim.x + threadIdx.x;
    if (i >= ET) return;
    int d = (i < N_EDGES) ? edst[i] : (i - N_EDGES);
    float ex = __expf(ebuf[i] - fl_unflip(mmax[d]));
    ebuf[i] = ex;
    atomicAdd(&z[d], ex);
}

// ================= K5: out[dst] += alpha * h[src]  (wave/edge) ==============
__global__ __launch_bounds__(256)
void gat_scatter(const int* __restrict__ esrc, const int* __restrict__ edst,
                 const float* __restrict__ ebuf, const float* __restrict__ z,
                 const float* __restrict__ h, float* __restrict__ out) {
    int e = (blockIdx.x * blockDim.x + threadIdx.x) >> 5;
    int lane = threadIdx.x & 31;
    if (e >= ET) return;
    int s, d;
    if (e < N_EDGES) { s = esrc[e]; d = edst[e]; } else { s = d = e - N_EDGES; }
    float alpha = ebuf[e] / (z[d] + 1e-16f);
    float2 hv = *(const float2*)(h + (size_t)s * HID + lane * 2);
    atomicAdd(&out[(size_t)d * HID + lane * 2 + 0], alpha * hv.x);
    atomicAdd(&out[(size_t)d * HID + lane * 2 + 1], alpha * hv.y);
}

// ================= K6: mean-pool accumulation (wave/node) ===================
__global__ __launch_bounds__(256)
void gat_pool(const float* __restrict__ out, const int* __restrict__ batch,
              float* __restrict__ pool, float* __restrict__ cnt) {
    int n = (blockIdx.x * blockDim.x + threadIdx.x) >> 5;
    int lane = threadIdx.x & 31;
    if (n >= N_NODES) return;
    int g = batch[n];
    float2 v = *(const float2*)(out + (size_t)n * HID + lane * 2);
    atomicAdd(&pool[(size_t)g * HID + lane * 2 + 0], v.x);
    atomicAdd(&pool[(size_t)g * HID + lane * 2 + 1], v.y);
    if (lane == 0) atomicAdd(&cnt[g], 1.0f);
}

// ================= K7: head: mean + bias, FC 64->3, log_softmax =============
__global__ __launch_bounds__(256)
void gat_head(const float* __restrict__ pool, const float* __restrict__ cnt,
              const float* __restrict__ bias_gat, const float* __restrict__ fc_w,
              const float* __restrict__ fc_b, float* __restrict__ y) {
    int g = blockIdx.x * blockDim.x + threadIdx.x;
    if (g >= N_GRAPHS) return;
    float inv = 1.0f / fmaxf(cnt[g], 1.0f);
    float lg[OUTC];
#pragma unroll
    for (int o = 0; o < OUTC; ++o) lg[o] = fc_b[o];
    for (int j = 0; j < HID; ++j) {
        float p = pool[(size_t)g * HID + j] * inv + bias_gat[j];
#pragma unroll
        for (int o = 0; o < OUTC; ++o) lg[o] += fc_w[o * HID + j] * p;
    }
    float m = fmaxf(lg[0], fmaxf(lg[1], lg[2]));
    float lse = logf(__expf(lg[0] - m) + __expf(lg[1] - m) + __expf(lg[2] - m));
#pragma unroll
    for (int o = 0; o < OUTC; ++o) y[g * OUTC + o] = lg[o] - m - lse;
}

// ============================ launcher ======================================
extern "C" void kernel_launch(void* const* d_in, const int* in_sizes, int n_in,
                              void* d_out, int out_size, void* d_ws, size_t ws_size,
                              hipStream_t stream) {
    const float* x        = (const float*)d_in[0];
    const int*   eidx     = (const int*)d_in[1];     // [2, E]: row0=src, row1=dst
    const int*   batch    = (const int*)d_in[2];
    const float* W        = (const float*)d_in[3];
    const float* att_src  = (const float*)d_in[4];
    const float* att_dst  = (const float*)d_in[5];
    const float* bias_gat = (const float*)d_in[6];
    const float* fc_w     = (const float*)d_in[7];
    const float* fc_b     = (const float*)d_in[8];
    float* y = (float*)d_out;

    float* ws = (float*)d_ws;
    float*        h    = ws + OFF_H;
    float*        a_s  = ws + OFF_AS;
    float*        a_d  = ws + OFF_AD;
    unsigned int* mmax = (unsigned int*)(ws + OFF_M);
    float*        z    = ws + OFF_Z;
    float*        ebuf = ws + OFF_E;
    float*        out  = ws + OFF_O;
    float*        pool = ws + OFF_P;
    float*        cnt  = ws + OFF_C;

    const int* esrc = eidx;
    const int* edst = eidx + N_EDGES;

    // zero the accumulator regions (max/z/ebuf/out/pool/cnt) in one memset node
    hipMemsetAsync((char*)d_ws + OFF_M * sizeof(float), 0,
                   (WS_FLOATS - OFF_M) * sizeof(float), stream);

    // K1: dense GEMM with f32 WMMA. 8192 row-tiles, 8 waves/block.
    gat_gemm_wmma<<<N_NODES / 16 / 8, 256, 0, stream>>>(x, W, h);

    // K2: attention dot products, one wave per node.
    gat_attn_dots<<<(N_NODES * 32) / 256, 256, 0, stream>>>(h, att_src, att_dst, a_s, a_d);

    // K3/K4: segment softmax over incoming edges (incl. self loops).
    gat_edge_logits<<<(ET + 255) / 256, 256, 0, stream>>>(esrc, edst, a_s, a_d, ebuf, mmax);
    gat_edge_exp<<<(ET + 255) / 256, 256, 0, stream>>>(edst, mmax, ebuf, z);

    // K5: weighted scatter-aggregate, one wave per edge.
    gat_scatter<<<(ET * 32 + 255) / 256, 256, 0, stream>>>(esrc, edst, ebuf, z, h, out);

    // K6: mean-pool accumulation, one wave per node.
    gat_pool<<<(N_NODES * 32) / 256, 256, 0, stream>>>(out, batch, pool, cnt);

    // K7: per-graph head.
    gat_head<<<(N_GRAPHS + 255) / 256, 256, 0, stream>>>(pool, cnt, bias_gat, fc_w, fc_b, y);
}